// ComponentIoULoss_5738076307612
// MI455X (gfx1250) — compile-verified
//
#include <hip/hip_runtime.h>
#include <hip/hip_bf16.h>
#include <math.h>

// Problem constants (from reference setup_inputs)
#define Bn   16
#define Hn   512
#define Wn   512
#define Nn   (Hn * Wn)        // 262144 = 1<<18, per-image pixel count
#define TOTn (Bn * Nn)        // 4194304
#define LOG2N 18
#define EPSf 0.05f
#define DILi 3
#define GLOBAL_ITERS 32
#define TILE 64
#define TILE_PX (TILE * TILE)  // 4096
#define LOCAL_ITERS 32
#define BLK 256

typedef __attribute__((ext_vector_type(2))) float v2f;
typedef __attribute__((ext_vector_type(8))) float v8f;

#if __has_builtin(__builtin_amdgcn_global_load_async_to_lds_b32) && \
    __has_builtin(__builtin_amdgcn_s_wait_asynccnt)
#define HAVE_ASYNC_LDS 1
typedef __attribute__((address_space(1))) int* gas_i32_ptr;
typedef __attribute__((address_space(3))) int* lds_i32_ptr;
#endif

// ---------------------------------------------------------------------------
// Wave-32 sum via V_WMMA_F32_16X16X4_F32 (layout reasoning: A[m][0]=p[m],
// A[m][2]=p[m+16], B=ones -> D[m][n]=p[m]+p[m+16]; lane sums of the 8 D VGPRs
// on lane0 (M=0..7) + lane16 (M=8..15) give the exact fp32 wave sum).
// EXEC must be all-1s at the call site (exact grid, no divergence).
// ---------------------------------------------------------------------------
__device__ __forceinline__ float wave_sum_f32(float s) {
#if __has_builtin(__builtin_amdgcn_wmma_f32_16x16x4_f32)
  v2f a; a[0] = s;   a[1] = 0.0f;
  v2f b; b[0] = 1.0f; b[1] = 1.0f;
  v8f c = {};
  c = __builtin_amdgcn_wmma_f32_16x16x4_f32(
      /*neg_a=*/false, a, /*neg_b=*/false, b,
      /*c_mod=*/(short)0, c, /*reuse_a=*/false, /*reuse_b=*/false);
  float t = c[0] + c[1] + c[2] + c[3] + c[4] + c[5] + c[6] + c[7];
  return __shfl(t, 0, 32) + __shfl(t, 16, 32);
#else
  float w = s;
  for (int off = 16; off > 0; off >>= 1) w += __shfl_down(w, off, 32);
  return __shfl(w, 0, 32);
#endif
}

__device__ __forceinline__ float block_sum(float s, float* lds8) {
  float wsum = wave_sum_f32(s);
  int lane = threadIdx.x & 31, wave = threadIdx.x >> 5;
  if (lane == 0) lds8[wave] = wsum;
  __syncthreads();
  float r = 0.0f;
  if (threadIdx.x == 0) {
    int nw = blockDim.x >> 5;
    for (int i = 0; i < nw; ++i) r += lds8[i];
  }
  __syncthreads();
  return r;  // valid on thread 0
}

// ---------------------------------------------------------------------------
// Kernel 0: zero scalar accumulators
// ---------------------------------------------------------------------------
__global__ void k_zero(double* bce, double* dice, int* rootCount) {
  if (threadIdx.x == 0) { *bce = 0.0; *dice = 0.0; *rootCount = 0; }
}

// ---------------------------------------------------------------------------
// Kernel 1: BCE partial sums + init bbox-stat arrays.
// loss(x,t) = log1p(exp(-|x|)) + (x>0 ? (1-t)*x : -t*x),  t = tgt*(1-2e)+e
// ---------------------------------------------------------------------------
__global__ void k_init(const float* __restrict__ logit,
                       const float* __restrict__ target,
                       int* __restrict__ rmax, int* __restrict__ cmin,
                       int* __restrict__ cmax, double* __restrict__ bce) {
  __shared__ float lds8[8];
  int i = blockIdx.x * BLK + threadIdx.x;  // exact grid, EXEC all-1s for WMMA
  float x  = logit[i];
  float tg = target[i];
  float t  = tg * (1.0f - 2.0f * EPSf) + EPSf;
  float ax = fabsf(x);
  float loss = log1pf(__expf(-ax)) + ((x > 0.0f) ? (1.0f - t) * x : -t * x);

  rmax[i] = -1;
  cmin[i] = Wn;
  cmax[i] = -1;

  float bs = block_sum(loss, lds8);
  if (threadIdx.x == 0) atomicAdd(bce, (double)bs);
}

// ---------------------------------------------------------------------------
// Kernel 2: tiled CCL pre-convergence. One block per 64x64 tile.
// Target tile is staged to LDS with GLOBAL_LOAD_ASYNC_TO_LDS_B32 (ASYNCcnt)
// when available; labels (global flat indices) converge locally via
// neighbor-min + 2 in-tile compression hops, then spill to global labels[].
// All label values produced here reference in-tile pixels, so compression is
// always locally resolvable. Monotone-decreasing -> races benign.
// ---------------------------------------------------------------------------
__global__ void k_ccl_tile(const float* __restrict__ target,
                           int* __restrict__ labels) {
  __shared__ int llab[TILE_PX];  // staged as f32 bits first, then labels
  int T = blockIdx.x;
  int img = T >> 6;              // 64 tiles per image (8x8)
  int tIn = T & 63;
  int tr = (tIn >> 3) * TILE;
  int tc = (tIn & 7) * TILE;
  int gbase = img * Nn;
  int tid = threadIdx.x;

#ifdef HAVE_ASYNC_LDS
  for (int k = 0; k < TILE_PX / BLK; ++k) {
    int q = tid + (k << 8);
    int lx = q & (TILE - 1), ly = q >> 6;
    const float* gp = target + gbase + ((tr + ly) << 9) + (tc + lx);
    int* gpi = (int*)const_cast<float*>(gp);   // generic int*
    int* lpi = &llab[q];                       // generic int* (LDS object)
    __builtin_amdgcn_global_load_async_to_lds_b32(
        (gas_i32_ptr)gpi, (lds_i32_ptr)lpi, 0, 0);
  }
  __builtin_amdgcn_s_wait_asynccnt(0);
  __syncthreads();
  // convert staged f32 bits -> labels (each thread converts what it loaded)
  for (int k = 0; k < TILE_PX / BLK; ++k) {
    int q = tid + (k << 8);
    int lx = q & (TILE - 1), ly = q >> 6;
    int p = ((tr + ly) << 9) | (tc + lx);
    float tg = __int_as_float(llab[q]);
    llab[q] = (tg > 0.5f) ? p : Nn;
  }
#else
  for (int k = 0; k < TILE_PX / BLK; ++k) {
    int q = tid + (k << 8);
    int lx = q & (TILE - 1), ly = q >> 6;
    int p = ((tr + ly) << 9) | (tc + lx);
    llab[q] = (target[gbase + p] > 0.5f) ? p : Nn;
  }
#endif
  __syncthreads();

  for (int it = 0; it < LOCAL_ITERS; ++it) {
    for (int k = 0; k < TILE_PX / BLK; ++k) {
      int q = tid + (k << 8);
      int l = llab[q];
      if (l == Nn) continue;
      int lx = q & (TILE - 1), ly = q >> 6;
      int m = l;
      if (lx > 0)        m = min(m, llab[q - 1]);
      if (lx < TILE - 1) m = min(m, llab[q + 1]);
      if (ly > 0)        m = min(m, llab[q - TILE]);
      if (ly < TILE - 1) m = min(m, llab[q + TILE]);
      // two in-tile path-compression hops (m always refers to in-tile pixel)
      int mq = (((m >> 9) - tr) << 6) | ((m & (Wn - 1)) - tc);
      m = llab[mq];
      mq = (((m >> 9) - tr) << 6) | ((m & (Wn - 1)) - tc);
      m = llab[mq];
      llab[q] = m;
    }
    __syncthreads();
  }

  for (int k = 0; k < TILE_PX / BLK; ++k) {
    int q = tid + (k << 8);
    int lx = q & (TILE - 1), ly = q >> 6;
    int p = ((tr + ly) << 9) | (tc + lx);
    labels[gbase + p] = llab[q];
  }
}

// ---------------------------------------------------------------------------
// Kernel 3: global CCL iteration (cross-tile merge): 4-neighbor min + two
// pointer-jump hops, in place. Monotone -> fixed point = component min index.
// ---------------------------------------------------------------------------
__global__ void k_prop(int* __restrict__ labels) {
  int i = blockIdx.x * BLK + threadIdx.x;
  int p = i & (Nn - 1);
  int* lab = labels + (i - p);  // image base
  int l = lab[p];
  if (l == Nn) return;          // background
  int r = p >> 9, c = p & (Wn - 1);
  int m = l;
  if (r > 0)      m = min(m, lab[p - Wn]);
  if (r < Hn - 1) m = min(m, lab[p + Wn]);
  if (c > 0)      m = min(m, lab[p - 1]);
  if (c < Wn - 1) m = min(m, lab[p + 1]);
  m = lab[m];   // label targets are foreground pixels -> lab[m] < Nn
  m = lab[m];
  lab[p] = m;
}

// ---------------------------------------------------------------------------
// Kernel 4: per-component bbox stats (rmin free: label>>9) + root compaction
// ---------------------------------------------------------------------------
__global__ void k_bbox(const int* __restrict__ labels,
                       int* __restrict__ rmax, int* __restrict__ cmin,
                       int* __restrict__ cmax, int* __restrict__ rootList,
                       int* __restrict__ rootCount) {
  int i = blockIdx.x * BLK + threadIdx.x;
  int l = labels[i];
  if (l == Nn) return;
  int p = i & (Nn - 1);
  int base = i - p;             // image base into stat arrays
  int r = p >> 9, c = p & (Wn - 1);
  atomicMax(&rmax[base + l], r);
  atomicMin(&cmin[base + l], c);
  atomicMax(&cmax[base + l], c);
  if (l == p) {                 // component root (min-index pixel)
    int pos = atomicAdd(rootCount, 1);
    if (pos < (1 << 18)) rootList[pos] = ((base >> LOG2N) << LOG2N) | p;
  }
}

// ---------------------------------------------------------------------------
// Kernel 5: per-component dilated-bbox dice. One block per component
// (grid-stride over compacted root list). prob recomputed as sigmoid(logit).
// ---------------------------------------------------------------------------
__global__ void k_dice(const float* __restrict__ logit,
                       const float* __restrict__ target,
                       const int* __restrict__ rmax, const int* __restrict__ cmin,
                       const int* __restrict__ cmax, const int* __restrict__ rootList,
                       const int* __restrict__ rootCount, double* __restrict__ dice) {
  __shared__ float sh[3 * BLK];
  int nRoots = *rootCount;
  if (nRoots > (1 << 18)) nRoots = (1 << 18);
  for (int e = blockIdx.x; e < nRoots; e += gridDim.x) {
    int packed = rootList[e];
    int b = packed >> LOG2N;
    int l = packed & (Nn - 1);
    int base = b * Nn;
    int rmin = l >> 9;
    int rmx = rmax[base + l], cmn = cmin[base + l], cmx = cmax[base + l];
    int r0 = max(rmin - DILi, 0), r1 = min(rmx + 1 + DILi, Hn);
    int c0 = max(cmn - DILi, 0),  c1 = min(cmx + 1 + DILi, Wn);
    int wdt = c1 - c0;
    int area = wdt * (r1 - r0);
    const float* lg = logit + base;
    const float* tg = target + base;
    float inter = 0.0f, ps = 0.0f, ms = 0.0f;
    for (int k = threadIdx.x; k < area; k += blockDim.x) {
      int r = r0 + k / wdt;
      int c = c0 + k % wdt;
      int q = r * Wn + c;
      float pr = 1.0f / (1.0f + __expf(-lg[q]));
      float m  = tg[q];
      inter += pr * m;
      ps    += pr;
      ms    += m;
    }
    int tid = threadIdx.x;
    sh[tid] = inter; sh[BLK + tid] = ps; sh[2 * BLK + tid] = ms;
    __syncthreads();
    for (int s = BLK / 2; s > 0; s >>= 1) {
      if (tid < s) {
        sh[tid]           += sh[tid + s];
        sh[BLK + tid]     += sh[BLK + tid + s];
        sh[2 * BLK + tid] += sh[2 * BLK + tid + s];
      }
      __syncthreads();
    }
    if (tid == 0) {
      float d = (2.0f * sh[0] + 1.0f) / (sh[BLK] + sh[2 * BLK] + 1.0f);
      atomicAdd(dice, (double)(1.0f - d));
    }
    __syncthreads();  // LDS reuse across grid-stride iterations
  }
}

// ---------------------------------------------------------------------------
// Kernel 6: finalize scalar loss
// ---------------------------------------------------------------------------
__global__ void k_final(const double* __restrict__ bce, const double* __restrict__ dice,
                        const int* __restrict__ rootCount, float* __restrict__ out) {
  if (threadIdx.x == 0) {
    double l_bce = *bce / (double)TOTn;
    int n = *rootCount;
    double l_comp = (n > 0) ? (*dice / (double)(n < 1 ? 1 : n)) : 0.0;
    out[0] = (float)(l_bce + l_comp);
  }
}

// ---------------------------------------------------------------------------
extern "C" void kernel_launch(void* const* d_in, const int* in_sizes, int n_in,
                              void* d_out, int out_size, void* d_ws, size_t ws_size,
                              hipStream_t stream) {
  const float* logit  = (const float*)d_in[0];
  const float* target = (const float*)d_in[1];
  float* out = (float*)d_out;

  // Workspace layout (~68.2 MB)
  char* ws = (char*)d_ws;
  size_t A = (size_t)TOTn * 4;
  int* labels   = (int*)(ws);
  int* rmax     = (int*)(ws + A);
  int* cmin     = (int*)(ws + 2 * A);
  int* cmax     = (int*)(ws + 3 * A);
  int* rootList = (int*)(ws + 4 * A);                 // 1<<18 ints = 1 MiB
  char* scal    = ws + 4 * A + ((size_t)1 << 20);
  double* bce   = (double*)scal;
  double* dice  = bce + 1;
  int* rootCount = (int*)(dice + 1);

  int nblk = TOTn / BLK;  // 16384

  k_zero<<<1, 32, 0, stream>>>(bce, dice, rootCount);
  k_init<<<nblk, BLK, 0, stream>>>(logit, target, rmax, cmin, cmax, bce);
  k_ccl_tile<<<TOTn / TILE_PX, BLK, 0, stream>>>(target, labels);  // 1024 tiles
  for (int it = 0; it < GLOBAL_ITERS; ++it)
    k_prop<<<nblk, BLK, 0, stream>>>(labels);
  k_bbox<<<nblk, BLK, 0, stream>>>(labels, rmax, cmin, cmax, rootList, rootCount);
  k_dice<<<2048, BLK, 0, stream>>>(logit, target, rmax, cmin, cmax, rootList,
                                   rootCount, dice);
  k_final<<<1, 32, 0, stream>>>(bce, dice, rootCount, out);
}